// LayerNormSubspace_40707700031573
// MI455X (gfx1250) — compile-verified
//
#include <hip/hip_runtime.h>

// LayerNorm + alpha-weighted subspace affine, organized around the CDNA5
// V_WMMA_F32_16X16X4_F32 layout so the mixture GEMM (alpha[16x4] @ W[4x16])
// runs on the matrix pipe in exact fp32 and its C-layout output is consumed
// element-for-element against x kept in registers (x read from HBM once).
//
// Shapes (fixed by the reference): B=4, S=8192 -> ROWS=32768, HS=1024, N=4.
// Bandwidth bound: ~256 MB of mandatory traffic -> ~11 us at 23.3 TB/s.

typedef float v2f __attribute__((ext_vector_type(2)));
typedef float v8f __attribute__((ext_vector_type(8)));

#define HS            1024
#define NANCH         4
#define ROWS_PER_BLK  16
#define LN_EPS        1e-5f

__global__ __launch_bounds__(256)
void ln_subspace_wmma(const float* __restrict__ x,
                      const float* __restrict__ alpha,
                      const float* __restrict__ W,
                      const float* __restrict__ Bb,
                      float* __restrict__ out) {
  // Cross-wave reduction scratch (tiny).
  __shared__ float sSum[8][16];
  __shared__ float sSq [8][16];
  __shared__ float sMean[16];
  __shared__ float sRstd[16];

  const int tid  = threadIdx.x;
  const int wave = tid >> 5;      // 0..7
  const int lane = tid & 31;      // wave32 lane
  const int half = lane >> 4;     // 0 or 1 (16-lane half)
  const int m16  = lane & 15;     // position within half

  const int row0 = blockIdx.x * ROWS_PER_BLK;
  const int colw = wave * 128 + m16;  // base column for this lane

  // ---------------- Phase 1: load x in WMMA C-layout, row stats ------------
  // xr[j][r] = x[row0 + r + 8*half][colw + 16*j]   (C layout: VGPR r <-> row)
  float xr[8][8];
  float ps[8], pq[8];
  #pragma unroll
  for (int r = 0; r < 8; ++r) { ps[r] = 0.f; pq[r] = 0.f; }

  #pragma unroll
  for (int r = 0; r < 8; ++r) {
    const size_t rbase = (size_t)(row0 + r + 8 * half) * HS + colw;
    #pragma unroll
    for (int j = 0; j < 8; ++j) {
      const float v = x[rbase + 16 * j];   // coalesced: 2x64B per instruction
      xr[j][r] = v;
      ps[r] += v;
      pq[r] += v * v;
    }
  }

  // Reduce over the 16 lanes of each half-wave (xor masks 1..8 stay in-half).
  #pragma unroll
  for (int r = 0; r < 8; ++r) {
    #pragma unroll
    for (int m = 1; m < 16; m <<= 1) {
      ps[r] += __shfl_xor(ps[r], m, 32);
      pq[r] += __shfl_xor(pq[r], m, 32);
    }
  }
  if (m16 == 0) {
    #pragma unroll
    for (int r = 0; r < 8; ++r) {
      sSum[wave][r + 8 * half] = ps[r];
      sSq [wave][r + 8 * half] = pq[r];
    }
  }
  __syncthreads();

  if (tid < 16) {
    float s = 0.f, q = 0.f;
    #pragma unroll
    for (int w = 0; w < 8; ++w) { s += sSum[w][tid]; q += sSq[w][tid]; }
    const float mean = s * (1.0f / HS);
    const float var  = q * (1.0f / HS) - mean * mean;  // biased, as reference
    sMean[tid] = mean;
    sRstd[tid] = rsqrtf(var + LN_EPS);
  }
  __syncthreads();

  float mr[8], rr[8];
  #pragma unroll
  for (int r = 0; r < 8; ++r) {
    mr[r] = sMean[r + 8 * half];
    rr[r] = sRstd[r + 8 * half];
  }

  // ---------------- Phase 2: WMMA mixture + fused normalize ----------------
  // A operand (16x4 f32): M = lane%16, K = vgpr + 2*half.
  v2f a;
  {
    const float2 av = *reinterpret_cast<const float2*>(
        alpha + (size_t)(row0 + m16) * NANCH + 2 * half);  // 8B aligned
    a.x = av.x;
    a.y = av.y;
  }

  #pragma unroll
  for (int j = 0; j < 8; ++j) {
    const int col = colw + 16 * j;

    // B operand (4x16 f32): row K = vgpr + 2*half, col = lane%16.
    v2f bw, bb;
    bw.x = W [(2 * half    ) * HS + col];
    bw.y = W [(2 * half + 1) * HS + col];
    bb.x = Bb[(2 * half    ) * HS + col];
    bb.y = Bb[(2 * half + 1) * HS + col];

    v8f cz = {};
    // D = A x B + 0 : exact fp32 matrix op on the WMMA pipe.
    v8f cw = __builtin_amdgcn_wmma_f32_16x16x4_f32(
        false, a, false, bw, (short)0, cz, false, false);
    v8f cb = __builtin_amdgcn_wmma_f32_16x16x4_f32(
        false, a, false, bb, (short)0, cz, false, false);

    // C-layout of cw/cb matches xr[j][*] exactly: VGPR r <-> row r + 8*half.
    #pragma unroll
    for (int r = 0; r < 8; ++r) {
      const float xh = (xr[j][r] - mr[r]) * rr[r];
      out[(size_t)(row0 + r + 8 * half) * HS + col] = xh * cw[r] + cb[r];
    }
  }
}

extern "C" void kernel_launch(void* const* d_in, const int* in_sizes, int n_in,
                              void* d_out, int out_size, void* d_ws, size_t ws_size,
                              hipStream_t stream) {
  (void)n_in; (void)out_size; (void)d_ws; (void)ws_size;
  const float* x     = (const float*)d_in[0];  // [B,S,HS]
  const float* alpha = (const float*)d_in[1];  // [B,S,N]
  const float* W     = (const float*)d_in[2];  // [N,HS]
  const float* b     = (const float*)d_in[3];  // [N,HS]
  float* out = (float*)d_out;

  const int rows = in_sizes[0] / HS;           // 32768
  const int grid = rows / ROWS_PER_BLK;        // 2048 blocks of 256 threads

  ln_subspace_wmma<<<grid, 256, 0, stream>>>(x, alpha, W, b, out);
}